// GraphCIW_27462020890936
// MI455X (gfx1250) — compile-verified
//
#include <hip/hip_runtime.h>

#define N_NODES 50000
#define N_EDGES 600000
#define D 128

typedef __attribute__((ext_vector_type(2))) float v2f;
typedef __attribute__((ext_vector_type(8))) float v8f;

// ---------------------------------------------------------------------------
// Degree count: one int atomic per edge.
// ---------------------------------------------------------------------------
__global__ void count_kernel(const long long* __restrict__ dst,
                             int* __restrict__ cnt, int E) {
    int e = blockIdx.x * blockDim.x + threadIdx.x;
    if (e < E) atomicAdd(&cnt[(int)dst[e]], 1);
}

// ---------------------------------------------------------------------------
// Single-block exclusive scan over n entries (n = 50000 -> 49 chunks of 1024).
// Cheap: runs once, reused by both layers.
// ---------------------------------------------------------------------------
__global__ __launch_bounds__(1024)
void scan_kernel(const int* __restrict__ cnt, int* __restrict__ rowstart, int n) {
    __shared__ int sdata[1024];
    __shared__ int carry;
    if (threadIdx.x == 0) carry = 0;
    __syncthreads();
    for (int base = 0; base < n; base += 1024) {
        int i = base + (int)threadIdx.x;
        int v = (i < n) ? cnt[i] : 0;
        sdata[threadIdx.x] = v;
        __syncthreads();
        for (int off = 1; off < 1024; off <<= 1) {
            int t = (threadIdx.x >= off) ? sdata[threadIdx.x - off] : 0;
            __syncthreads();
            sdata[threadIdx.x] += t;
            __syncthreads();
        }
        if (i < n) rowstart[i] = carry + sdata[threadIdx.x] - v;  // exclusive
        __syncthreads();
        if (threadIdx.x == 1023) carry += sdata[1023];
        __syncthreads();
    }
}

// ---------------------------------------------------------------------------
// CSR fill: place src index into its dst bucket (600k int atomics).
// ---------------------------------------------------------------------------
__global__ void fill_csr_kernel(const long long* __restrict__ src,
                                const long long* __restrict__ dst,
                                int* __restrict__ cursor,
                                int* __restrict__ csr_src, int E) {
    int e = blockIdx.x * blockDim.x + threadIdx.x;
    if (e >= E) return;
    int d = (int)dst[e];
    int pos = atomicAdd(&cursor[d], 1);
    csr_src[pos] = (int)src[e];
}

// ---------------------------------------------------------------------------
// Gather-based mean aggregation: one wave per node. Each lane owns a float4
// slice of the 128-float row; neighbor rows are streamed with coalesced 512B
// gathers and summed in registers. No fp atomics anywhere.
// ---------------------------------------------------------------------------
__global__ __launch_bounds__(256)
void aggregate_kernel(const float* __restrict__ x,
                      const int* __restrict__ csr_src,
                      const int* __restrict__ rowstart,
                      const int* __restrict__ cnt,
                      float* __restrict__ mean) {
    int node = blockIdx.x * 8 + (threadIdx.x >> 5);   // 8 waves per block
    int lane = threadIdx.x & 31;
    if (node >= N_NODES) return;
    int beg = rowstart[node];
    int deg = cnt[node];
    float4 acc = make_float4(0.f, 0.f, 0.f, 0.f);
    for (int i = 0; i < deg; ++i) {
        int s = csr_src[beg + i];                      // uniform per wave
        float4 v = *((const float4*)(x + (size_t)s * D) + lane);
        acc.x += v.x; acc.y += v.y; acc.z += v.z; acc.w += v.w;
    }
    float sc = 1.0f / fmaxf((float)deg, 1.0f);
    acc.x *= sc; acc.y *= sc; acc.z *= sc; acc.w *= sc;
    *((float4*)(mean + (size_t)node * D) + lane) = acc;
}

// ---------------------------------------------------------------------------
// Fused SAGEConv GEMM: out[16 rows x 128 cols per block] =
//     mean @ Wl + b + x @ Wr   [+ relu]
// 8 waves per block, each wave owns one 16x16 tile via V_WMMA_F32_16X16X4_F32.
// ---------------------------------------------------------------------------
__global__ __launch_bounds__(256)
void sage_gemm_kernel(const float* __restrict__ mean,
                      const float* __restrict__ xin,
                      const float* __restrict__ Wl,
                      const float* __restrict__ Wr,
                      const float* __restrict__ bias,
                      float* __restrict__ out, int relu) {
    __shared__ float ldsMean[16 * D];
    __shared__ float ldsX[16 * D];

    const int row0 = blockIdx.x * 16;
    const int tid = threadIdx.x;

    // Stage 16x128 mean-tile and x-tile (2048 floats each): 2 float4 / thread.
#pragma unroll
    for (int i = 0; i < 2; ++i) {
        int idx = (tid + i * 256) * 4;      // flat float index [0, 2048)
        int r = idx >> 7;
        int k = idx & (D - 1);
        *(float4*)(ldsX + r * D + k) =
            *(const float4*)(xin + (size_t)(row0 + r) * D + k);
        *(float4*)(ldsMean + r * D + k) =
            *(const float4*)(mean + (size_t)(row0 + r) * D + k);
    }
    __syncthreads();

    const int wave  = tid >> 5;         // 0..7 -> N tile
    const int lane  = tid & 31;
    const int n0    = wave * 16;
    const int mrow  = lane & 15;        // M index (A) / N index (B)
    const int khalf = (lane >> 4) * 2;  // 0 or 2 (K sub-offset per ISA layout)

    v8f acc = {};

    // mean @ Wl  (K = 128, step 4)
#pragma unroll 4
    for (int k0 = 0; k0 < D; k0 += 4) {
        v2f a, b;
        const float* ap = ldsMean + mrow * D + k0 + khalf;
        a.x = ap[0];
        a.y = ap[1];
        const float* bp = Wl + (size_t)(k0 + khalf) * D + n0 + mrow;
        b.x = bp[0];
        b.y = bp[D];
        acc = __builtin_amdgcn_wmma_f32_16x16x4_f32(false, a, false, b,
                                                    (short)0, acc, false, false);
    }
    // x @ Wr
#pragma unroll 4
    for (int k0 = 0; k0 < D; k0 += 4) {
        v2f a, b;
        const float* ap = ldsX + mrow * D + k0 + khalf;
        a.x = ap[0];
        a.y = ap[1];
        const float* bp = Wr + (size_t)(k0 + khalf) * D + n0 + mrow;
        b.x = bp[0];
        b.y = bp[D];
        acc = __builtin_amdgcn_wmma_f32_16x16x4_f32(false, a, false, b,
                                                    (short)0, acc, false, false);
    }

    // C/D layout: VGPR v -> M = v (lanes 0-15) or v+8 (lanes 16-31), N = lane%16.
    const int ncol = n0 + mrow;
    const float bv = bias[ncol];
    const int mbase = row0 + ((lane >> 4) * 8);
#pragma unroll
    for (int v = 0; v < 8; ++v) {
        float r = acc[v] + bv;
        if (relu) r = fmaxf(r, 0.0f);
        out[(size_t)(mbase + v) * D + ncol] = r;
    }
}

// ---------------------------------------------------------------------------
extern "C" void kernel_launch(void* const* d_in, const int* in_sizes, int n_in,
                              void* d_out, int out_size, void* d_ws, size_t ws_size,
                              hipStream_t stream) {
    const float*     x   = (const float*)d_in[0];
    const long long* ei  = (const long long*)d_in[1];   // int64 [2, E]
    const float*     W1l = (const float*)d_in[2];
    const float*     b1  = (const float*)d_in[3];
    const float*     W1r = (const float*)d_in[4];
    const float*     W2l = (const float*)d_in[5];
    const float*     b2  = (const float*)d_in[6];
    const float*     W2r = (const float*)d_in[7];
    float*           out = (float*)d_out;

    const long long* src = ei;
    const long long* dst = ei + N_EDGES;

    // Workspace layout (floats first for 16B alignment of float4 rows)
    float* mean    = (float*)d_ws;                         // N*128
    float* h1      = mean + (size_t)N_NODES * D;           // N*128
    int*   cnt     = (int*)(h1 + (size_t)N_NODES * D);     // N
    int*   rowstart= cnt + N_NODES;                        // N
    int*   cursor  = rowstart + N_NODES;                   // N
    int*   csr_src = cursor + N_NODES;                     // E

    dim3 blk(256);
    dim3 gridEdges((N_EDGES + 255) / 256);
    dim3 gridNodes((N_NODES + 7) / 8);     // wave per node, 8 waves/block
    dim3 gridGemm(N_NODES / 16);           // 3125 blocks, exact

    // ---- Build CSR (once, shared by both layers) ----
    hipMemsetAsync(cnt, 0, (size_t)N_NODES * sizeof(int), stream);
    count_kernel<<<gridEdges, blk, 0, stream>>>(dst, cnt, N_EDGES);
    scan_kernel<<<1, 1024, 0, stream>>>(cnt, rowstart, N_NODES);
    hipMemcpyAsync(cursor, rowstart, (size_t)N_NODES * sizeof(int),
                   hipMemcpyDeviceToDevice, stream);
    fill_csr_kernel<<<gridEdges, blk, 0, stream>>>(src, dst, cursor, csr_src,
                                                   N_EDGES);

    // ---- Layer 1 ----
    aggregate_kernel<<<gridNodes, blk, 0, stream>>>(x, csr_src, rowstart, cnt,
                                                    mean);
    sage_gemm_kernel<<<gridGemm, blk, 0, stream>>>(mean, x, W1l, W1r, b1,
                                                   h1, /*relu=*/1);

    // ---- Layer 2 ----
    aggregate_kernel<<<gridNodes, blk, 0, stream>>>(h1, csr_src, rowstart, cnt,
                                                    mean);
    sage_gemm_kernel<<<gridGemm, blk, 0, stream>>>(mean, h1, W2l, W2r, b2,
                                                   out, /*relu=*/0);
}